// ConductanceLIF_35424890258154
// MI455X (gfx1250) — compile-verified
//
#include <hip/hip_runtime.h>

// ---------------------------------------------------------------------------
// Conductance-LIF scan, T=512 steps x N=65536 neurons.
// Memory-bound streaming kernel: async global->LDS (b128) triple-buffered
// staging on the CDNA5 ASYNCcnt path, wave32, 4 neurons per thread,
// no block barriers (each thread consumes only its own async-loaded LDS).
// ---------------------------------------------------------------------------

#define BLOCK   64                  // threads per block
#define VEC     4                   // neurons per thread (b128 granularity)
#define CHUNK   8                   // timesteps staged per LDS buffer
#define NBUF    3                   // triple buffer: 2 chunks always in flight
#define TSTEPS  512
#define NNEUR   65536
#define NGRP    (NNEUR / VEC)       // 16384 float4 groups per row
#define NCH     (TSTEPS / CHUNK)    // 64 chunks

typedef float v4f __attribute__((ext_vector_type(4)));
typedef int   v4i __attribute__((vector_size(16)));   // matches builtin param

// ---- CDNA5 async global->LDS support --------------------------------------
#if defined(__gfx1250__) && __has_builtin(__builtin_amdgcn_global_load_async_to_lds_b128)
#define HAVE_ASYNC128 1
#else
#define HAVE_ASYNC128 0
#endif
#if defined(__gfx1250__) && __has_builtin(__builtin_amdgcn_global_load_async_to_lds_b32)
#define HAVE_ASYNC32 1
#else
#define HAVE_ASYNC32 0
#endif

#if __has_builtin(__builtin_amdgcn_s_wait_asynccnt)
#define WAIT_ASYNC(n) __builtin_amdgcn_s_wait_asynccnt(n)
#elif defined(__gfx1250__)
#define WAIT_ASYNC(n) asm volatile("s_wait_asynccnt " #n ::: "memory")
#else
#define WAIT_ASYNC(n)
#endif

typedef __attribute__((address_space(1))) int gas_int;   // global AS scalar
typedef __attribute__((address_space(3))) int las_int;   // LDS AS scalar
typedef __attribute__((address_space(1))) v4i gas_v4i;   // global AS vector
typedef __attribute__((address_space(3))) v4i las_v4i;   // LDS AS vector

// Copy 16 bytes (one float4) global -> LDS via the async data mover.
__device__ __forceinline__ void async_copy_v4(const v4f* g, v4f* l) {
#if HAVE_ASYNC128
    // cpol = 1 -> TH_NT: inputs are stream-once (384MB >> 192MB L2)
    __builtin_amdgcn_global_load_async_to_lds_b128((gas_v4i*)g, (las_v4i*)l,
                                                   /*offset=*/0, /*cpol=*/1);
#elif HAVE_ASYNC32
    const float* gf = (const float*)g;
    float*       lf = (float*)l;
    #pragma unroll
    for (int k = 0; k < 4; ++k)
        __builtin_amdgcn_global_load_async_to_lds_b32((gas_int*)(gf + k),
                                                      (las_int*)(lf + k), 0, 1);
#else
    *l = __builtin_nontemporal_load(g);   // fallback: synchronous copy
#endif
}

// Stage one chunk (3 streams x CHUNK rows) for this thread's 4-neuron column.
__device__ __forceinline__ void stage_chunk(const v4f* __restrict__ s0,
                                            const v4f* __restrict__ s1,
                                            const v4f* __restrict__ s2,
                                            int baseGrp,
                                            v4f (*__restrict__ dst)[CHUNK][BLOCK],
                                            int tid) {
    #pragma unroll
    for (int tl = 0; tl < CHUNK; ++tl) {
        async_copy_v4(s0 + baseGrp + tl * NGRP, &dst[0][tl][tid]);
        async_copy_v4(s1 + baseGrp + tl * NGRP, &dst[1][tl][tid]);
        async_copy_v4(s2 + baseGrp + tl * NGRP, &dst[2][tl][tid]);
    }
}

__global__ __launch_bounds__(BLOCK) void ConductanceLIF_35424890258154_kernel(
    const float* __restrict__ g_exc,
    const float* __restrict__ g_inh,
    const float* __restrict__ noise,
    const float* __restrict__ v_threshold,
    const float* __restrict__ tau_ref,
    float* __restrict__ out_spk,
    float* __restrict__ out_mem)
{
    // [buffer][stream][t_local][lane]; 3*3*8*64*16B = 72 KB
    __shared__ v4f lds[NBUF][3][CHUNK][BLOCK];

    const int tid  = threadIdx.x;
    const int grp  = blockIdx.x * BLOCK + tid;       // float4-group index

    const v4f* s0 = (const v4f*)g_exc;
    const v4f* s1 = (const v4f*)g_inh;
    const v4f* s2 = (const v4f*)noise;

    // ---- prologue: stage chunks 0 and 1 (2*24 = 48 async b128 loads) ----
    stage_chunk(s0, s1, s2, 0 * CHUNK * NGRP + grp, lds[0], tid);
    stage_chunk(s0, s1, s2, 1 * CHUNK * NGRP + grp, lds[1], tid);

    const v4f vthr = ((const v4f*)v_threshold)[grp];
    const v4f tref = ((const v4f*)tau_ref)[grp];

    v4f v   = {0.f, 0.f, 0.f, 0.f};
    v4f gE  = v, gI = v, ou = v, ref = v;
    // gA is identically zero: ADAPT_INC == 0 and gA(0) == 0 -> eliminated.

    constexpr float GE_DECAY = 0.8187307530779818f;   // exp(-1/5)
    constexpr float GI_DECAY = 0.9048374180359595f;   // exp(-1/10)
    constexpr float OU_ALPHA = 0.8187307530779818f;   // exp(-1/5)
    constexpr float OU_SIGMA = 0.011483553000933251f; // 0.02*sqrt(1-alpha^2)
    constexpr float NEG_K_LOG2E = -0.05f * 1.4426950408889634f; // exp -> exp2

    v4f* ospk = (v4f*)out_spk;
    v4f* omem = (v4f*)out_mem;

    int buf = 0;                       // buffer holding chunk c
    for (int c = 0; c < NCH; ++c) {
        if (c + 2 < NCH) {
            // prefetch chunk c+2 into the rotating buffer two ahead
            int b2 = buf + 2; if (b2 >= NBUF) b2 -= NBUF;
            stage_chunk(s0, s1, s2, (c + 2) * CHUNK * NGRP + grp, lds[b2], tid);
            // in-order completion: <=48 outstanding => chunk c has landed
            WAIT_ASYNC(48);
        } else if (c + 1 < NCH) {
            WAIT_ASYNC(24);
        } else {
            WAIT_ASYNC(0);
        }

        #pragma unroll
        for (int tl = 0; tl < CHUNK; ++tl) {
            const v4f ge4 = lds[buf][0][tl][tid];   // ds_load_b128
            const v4f gi4 = lds[buf][1][tl][tid];
            const v4f z4  = lds[buf][2][tl][tid];

            v4f spk4;
            #pragma unroll
            for (int k = 0; k < VEC; ++k) {         // 4 independent recurrences
                float e = fmaf(gE[k], GE_DECAY, ge4[k]);
                float i = fmaf(gI[k], GI_DECAY, gi4[k]);
                gE[k] = e; gI[k] = i;

                const float g_tot = 1.0f + e + i;                 // / g_L
                const float v_inf = (3.0f * e - 0.5f * i) *
                                    __builtin_amdgcn_rcpf(g_tot); // E_L = 0
                const float decay = __builtin_amdgcn_exp2f(NEG_K_LOG2E * g_tot);
                float vv = fmaf(v[k] - v_inf, decay, v_inf);

                const float o = fmaf(ou[k], OU_ALPHA, OU_SIGMA * z4[k]);
                ou[k] = o;
                vv += o;

                const bool in_ref = ref[k] > 0.0f;
                vv = in_ref ? 0.0f : vv;
                const bool spike = (!in_ref) && (vv >= vthr[k]);
                vv = spike ? 0.0f : vv;
                ref[k] = spike ? tref[k] : fmaxf(ref[k] - 1.0f, 0.0f);

                v[k]    = vv;
                spk4[k] = spike ? 1.0f : 0.0f;
            }

            const int idx = (c * CHUNK + tl) * NGRP + grp;
            __builtin_nontemporal_store(spk4, ospk + idx);   // b128 NT store
            __builtin_nontemporal_store(v,    omem + idx);   // b128 NT store
        }

        if (++buf >= NBUF) buf = 0;
    }
}

extern "C" void kernel_launch(void* const* d_in, const int* in_sizes, int n_in,
                              void* d_out, int out_size, void* d_ws, size_t ws_size,
                              hipStream_t stream) {
    (void)in_sizes; (void)n_in; (void)out_size; (void)d_ws; (void)ws_size;

    const float* g_exc = (const float*)d_in[0];
    const float* g_inh = (const float*)d_in[1];
    const float* noise = (const float*)d_in[2];
    const float* vthr  = (const float*)d_in[3];
    const float* tref  = (const float*)d_in[4];

    float* out     = (float*)d_out;
    float* out_spk = out;                                   // spikes [T,N]
    float* out_mem = out + (size_t)TSTEPS * (size_t)NNEUR;  // membranes [T,N]

    dim3 grid(NGRP / BLOCK);   // 256 blocks
    dim3 block(BLOCK);
    hipLaunchKernelGGL(ConductanceLIF_35424890258154_kernel, grid, block, 0, stream,
                       g_exc, g_inh, noise, vthr, tref, out_spk, out_mem);
}